// MTGCNNet_17231408792162
// MI455X (gfx1250) — compile-verified
//
#include <hip/hip_runtime.h>

typedef __attribute__((ext_vector_type(2))) float v2f;
typedef __attribute__((ext_vector_type(8))) float v8f;

#define DIN 58
#define H1  300
#define H2  100
// padded dims (multiples of 16)
#define KP0 64    // padded DIN
#define NP1 304   // padded H1
#define NP2 112   // padded H2

// ---------------- utility kernels ----------------

__global__ void k_zero(float* __restrict__ p, long n) {
  long i = (long)blockIdx.x * blockDim.x + threadIdx.x;
  long stride = (long)gridDim.x * blockDim.x;
  for (; i < n; i += stride) p[i] = 0.0f;
}

// out (Kp x Np) = zero-padded copy of in (K x Nn)
__global__ void k_padmat(const float* __restrict__ in, float* __restrict__ out,
                         int K, int Nn, int Kp, int Np) {
  int total = Kp * Np;
  for (int idx = blockIdx.x * blockDim.x + threadIdx.x; idx < total;
       idx += gridDim.x * blockDim.x) {
    int k = idx / Np, n = idx - k * Np;
    out[idx] = (k < K && n < Nn) ? in[k * Nn + n] : 0.0f;
  }
}

// out (M x Kp) = zero-padded copy of in (M x K)
__global__ void k_padrows(const float* __restrict__ in, float* __restrict__ out,
                          long M, int K, int Kp) {
  long total = M * (long)Kp;
  long stride = (long)gridDim.x * blockDim.x;
  for (long idx = (long)blockIdx.x * blockDim.x + threadIdx.x; idx < total; idx += stride) {
    long r = idx / Kp;
    int c = (int)(idx - r * Kp);
    out[idx] = (c < K) ? in[r * (long)K + c] : 0.0f;
  }
}

__global__ void k_deg(const int* __restrict__ row, float* __restrict__ deg, int E) {
  int e = blockIdx.x * blockDim.x + threadIdx.x;
  if (e < E) atomicAdd(&deg[row[e]], 1.0f);
}

__global__ void k_dinv(float* __restrict__ deg, int n) {
  int i = blockIdx.x * blockDim.x + threadIdx.x;
  if (i < n) {
    float d = deg[i];
    deg[i] = (d > 0.0f) ? rsqrtf(fmaxf(d, 1.0f)) : 0.0f;
  }
}

__global__ void k_edgew(const int* __restrict__ row, const int* __restrict__ col,
                        const float* __restrict__ dinv, float* __restrict__ w, int E) {
  int e = blockIdx.x * blockDim.x + threadIdx.x;
  if (e < E) w[e] = -dinv[row[e]] * dinv[col[e]];
}

// tx1[dst] += w[e] * X[src], 4 features per thread (padded cols are zero -> harmless)
__global__ void k_scatter4(const float* __restrict__ X, const int* __restrict__ src,
                           const int* __restrict__ dst, const float* __restrict__ w,
                           float* __restrict__ out, long E, int F4, int sX, int sO) {
  long idx = (long)blockIdx.x * blockDim.x + threadIdx.x;
  long total = E * (long)F4;
  long stride = (long)gridDim.x * blockDim.x;
  for (; idx < total; idx += stride) {
    long e = idx / F4;
    int j = (int)(idx - e * (long)F4) << 2;
    float we = w[e];
    const float4 v = *(const float4*)(X + (long)src[e] * sX + j);
    float* o = out + (long)dst[e] * sO + j;
    atomicAdd(o + 0, we * v.x);
    atomicAdd(o + 1, we * v.y);
    atomicAdd(o + 2, we * v.z);
    atomicAdd(o + 3, we * v.w);
  }
}

// ---------------- WMMA fp32 GEMM (padded, no bounds checks) ----------------
// Block: 256 threads = 8 waves, computes 128 rows x 16 cols.
// B tile(s) staged transposed in LDS: Bs[col*KP + k], so (b.x,b.y) = ds_load_b64.

template <int NA, bool RESID, int KP>
__global__ __launch_bounds__(256) void k_gemm(
    const float* __restrict__ A0, const float* __restrict__ A1,
    const float* __restrict__ W0, const float* __restrict__ W1,
    const float* __restrict__ bias, const float* __restrict__ resid,
    float* __restrict__ D, int M, int Np) {
  __shared__ float Bs[NA * KP * 16];
  float* Bs0 = Bs;
  float* Bs1 = Bs + KP * 16;
  const int n0 = blockIdx.x << 4;

  // cooperative transpose-stage of weight tiles (coalesced 64B global reads)
  for (int idx = threadIdx.x; idx < (KP << 4); idx += 256) {
    int col = idx & 15, k = idx >> 4;
    Bs0[col * KP + k] = W0[(long)k * Np + n0 + col];
    if (NA == 2) Bs1[col * KP + k] = W1[(long)k * Np + n0 + col];
  }
  __syncthreads();

  const int m0 = (blockIdx.y << 7) + ((threadIdx.x >> 5) << 4);
  if (m0 >= M) return;  // wave-uniform; EXEC all-1s below
  const int lane = threadIdx.x & 31;
  const int rc = lane & 15;
  const int kh = (lane >> 4) << 1;  // 0 | 2

  const float* a0p = A0 + (long)(m0 + rc) * KP + kh;
  const float* a1p = (NA == 2) ? (A1 + (long)(m0 + rc) * KP + kh) : a0p;
  const float* b0p = Bs0 + rc * KP + kh;
  const float* b1p = Bs1 + rc * KP + kh;

  v8f c = {0.f, 0.f, 0.f, 0.f, 0.f, 0.f, 0.f, 0.f};
#pragma unroll 4
  for (int k = 0; k < KP; k += 4) {
    v2f a = *(const v2f*)(a0p + k);
    v2f b = *(const v2f*)(b0p + k);
    c = __builtin_amdgcn_wmma_f32_16x16x4_f32(false, a, false, b, (short)0, c, false, false);
    if (NA == 2) {
      v2f a1 = *(const v2f*)(a1p + k);
      v2f b1 = *(const v2f*)(b1p + k);
      c = __builtin_amdgcn_wmma_f32_16x16x4_f32(false, a1, false, b1, (short)0, c, false, false);
    }
  }

  const int col = n0 + rc;
  const float bv = bias[col];
  const int rbase = m0 + ((lane >> 4) << 3);
#pragma unroll
  for (int v = 0; v < 8; ++v) {
    long o = (long)(rbase + v) * Np + col;
    float val = fmaxf(c[v] + bv, 0.0f);
    if (RESID) val += resid[o];
    D[o] = val;
  }
}

// ---------------- link loss ----------------

__global__ void k_loss(const float* __restrict__ z, const int* __restrict__ a,
                       const int* __restrict__ b, float* __restrict__ accum,
                       int E, int F4, int sZ, int neg) {
  int e = blockIdx.x * blockDim.x + threadIdx.x;
  float v = 0.0f;
  if (e < E) {
    const float* za = z + (long)a[e] * sZ;
    const float* zb = z + (long)b[e] * sZ;
    float dot = 0.0f;
    for (int f = 0; f < F4; ++f) {
      float4 u = *(const float4*)(za + 4 * f);
      float4 t = *(const float4*)(zb + 4 * f);
      dot += u.x * t.x + u.y * t.y + u.z * t.z + u.w * t.w;
    }
    float s = 1.0f / (1.0f + expf(-dot));
    float p = neg ? (1.0f - s) : s;
    v = -logf(p + 1e-15f);
  }
  for (int off = 16; off > 0; off >>= 1) v += __shfl_down(v, off, 32);
  if ((threadIdx.x & 31) == 0) atomicAdd(&accum[neg], v);
}

// ---------------- conv3 (H2 -> 1) + finalize ----------------

__global__ void k_conv3(const float* __restrict__ xs, const float* __restrict__ tx,
                        const float* __restrict__ w0, const float* __restrict__ w1,
                        const float* __restrict__ b, float* __restrict__ out,
                        int N, int F4, int sX) {
  int i = blockIdx.x * blockDim.x + threadIdx.x;
  if (i >= N) return;
  float acc = b[0];
  const float* p0 = xs + (long)i * sX;
  const float* p1 = tx + (long)i * sX;
  for (int k = 0; k < F4; ++k) {
    float4 u0 = *(const float4*)(p0 + 4 * k);
    float4 u1 = *(const float4*)(p1 + 4 * k);
    float4 a = *(const float4*)(w0 + 4 * k);
    float4 bb = *(const float4*)(w1 + 4 * k);
    acc += u0.x * a.x + u0.y * a.y + u0.z * a.z + u0.w * a.w;
    acc += u1.x * bb.x + u1.y * bb.y + u1.z * bb.z + u1.w * bb.w;
  }
  out[i] = acc;
}

__global__ void k_final(const float* __restrict__ accum, const float* __restrict__ c1,
                        const float* __restrict__ c2, float* __restrict__ out,
                        int N, float invE) {
  if (threadIdx.x == 0 && blockIdx.x == 0) {
    out[N]     = accum[0] * invE + accum[1] * invE;
    out[N + 1] = c1[0];
    out[N + 2] = c2[0];
  }
}

// ---------------- launch ----------------

static inline int cdiv(long a, long b) { return (int)((a + b - 1) / b); }

extern "C" void kernel_launch(void* const* d_in, const int* in_sizes, int n_in,
                              void* d_out, int out_size, void* d_ws, size_t ws_size,
                              hipStream_t stream) {
  const float* x    = (const float*)d_in[0];
  const int*   ei   = (const int*)d_in[1];
  const int*   nei  = (const int*)d_in[2];
  const float* c1w0 = (const float*)d_in[3];
  const float* c1w1 = (const float*)d_in[4];
  const float* c1b  = (const float*)d_in[5];
  const float* c2w0 = (const float*)d_in[6];
  const float* c2w1 = (const float*)d_in[7];
  const float* c2b  = (const float*)d_in[8];
  const float* c3w0 = (const float*)d_in[9];
  const float* c3w1 = (const float*)d_in[10];
  const float* c3b  = (const float*)d_in[11];
  const float* l1w  = (const float*)d_in[12];
  const float* l1b  = (const float*)d_in[13];
  const float* l2w  = (const float*)d_in[14];
  const float* l2b  = (const float*)d_in[15];
  const float* c1s  = (const float*)d_in[16];
  const float* c2s  = (const float*)d_in[17];

  const int N = in_sizes[0] / DIN;   // 100000, multiple of 16
  const int E = in_sizes[1] / 2;
  const int* src = ei;
  const int* dst = ei + E;

  float* out = (float*)d_out;
  float* ws  = (float*)d_ws;

  // workspace layout (floats)
  long off = 0;
  float* deg   = ws + off; off += N;
  float* wE    = ws + off; off += E;
  float* xpad  = ws + off; off += (long)N * KP0;   // N x 64
  float* bufT  = ws + off; off += (long)N * NP1;   // tx buffers (stride 64/304/112)
  float* bufH  = ws + off; off += (long)N * NP1;   // h, N x 304
  float* x1    = ws + off; off += (long)N * NP2;   // N x 112
  float* xs    = ws + off; off += (long)N * NP2;
  float* zz    = ws + off; off += (long)N * NP2;
  float* accum = ws + off; off += 4;
  float* w0p1  = ws + off; off += KP0 * NP1;
  float* w1p1  = ws + off; off += KP0 * NP1;
  float* b1p   = ws + off; off += NP1;
  float* w0p2  = ws + off; off += NP1 * NP2;
  float* w1p2  = ws + off; off += NP1 * NP2;
  float* b2p   = ws + off; off += NP2;
  float* l1wp  = ws + off; off += KP0 * NP2;
  float* l1bp  = ws + off; off += NP2;
  float* l2wp  = ws + off; off += KP0 * NP2;
  float* l2bp  = ws + off; off += NP2;

  const int TB = 256;
  const int ZB = 2048;

  // 0) padded weight/bias/input copies
  k_padmat<<<cdiv(KP0 * NP1, TB), TB, 0, stream>>>(c1w0, w0p1, DIN, H1, KP0, NP1);
  k_padmat<<<cdiv(KP0 * NP1, TB), TB, 0, stream>>>(c1w1, w1p1, DIN, H1, KP0, NP1);
  k_padmat<<<cdiv(NP1, TB), TB, 0, stream>>>(c1b, b1p, 1, H1, 1, NP1);
  k_padmat<<<cdiv(NP1 * NP2, TB), TB, 0, stream>>>(c2w0, w0p2, H1, H2, NP1, NP2);
  k_padmat<<<cdiv(NP1 * NP2, TB), TB, 0, stream>>>(c2w1, w1p2, H1, H2, NP1, NP2);
  k_padmat<<<cdiv(NP2, TB), TB, 0, stream>>>(c2b, b2p, 1, H2, 1, NP2);
  k_padmat<<<cdiv(KP0 * NP2, TB), TB, 0, stream>>>(l1w, l1wp, DIN, H2, KP0, NP2);
  k_padmat<<<cdiv(NP2, TB), TB, 0, stream>>>(l1b, l1bp, 1, H2, 1, NP2);
  k_padmat<<<cdiv(KP0 * NP2, TB), TB, 0, stream>>>(l2w, l2wp, DIN, H2, KP0, NP2);
  k_padmat<<<cdiv(NP2, TB), TB, 0, stream>>>(l2b, l2bp, 1, H2, 1, NP2);
  k_padrows<<<ZB, TB, 0, stream>>>(x, xpad, (long)N, DIN, KP0);

  // 1) normalization weights
  k_zero<<<ZB, TB, 0, stream>>>(deg, (long)N);
  k_zero<<<1, 64, 0, stream>>>(accum, 4L);
  k_deg<<<cdiv(E, TB), TB, 0, stream>>>(src, deg, E);
  k_dinv<<<cdiv(N, TB), TB, 0, stream>>>(deg, N);
  k_edgew<<<cdiv(E, TB), TB, 0, stream>>>(src, dst, deg, wE, E);

  // 2) conv1: h = relu(x@W0 + tx1@W1 + b)
  k_zero<<<ZB, TB, 0, stream>>>(bufT, (long)N * KP0);
  k_scatter4<<<4096, TB, 0, stream>>>(xpad, src, dst, wE, bufT, (long)E, 60 / 4, KP0, KP0);
  {
    dim3 g(NP1 / 16, cdiv(N, 128));
    k_gemm<2, false, KP0><<<g, TB, 0, stream>>>(xpad, bufT, w0p1, w1p1, b1p, nullptr,
                                                bufH, N, NP1);
  }

  // 3) conv2: x1 = relu(h@W0 + tx1(h)@W1 + b)
  k_zero<<<ZB, TB, 0, stream>>>(bufT, (long)N * NP1);
  k_scatter4<<<8192, TB, 0, stream>>>(bufH, src, dst, wE, bufT, (long)E, H1 / 4, NP1, NP1);
  {
    dim3 g(NP2 / 16, cdiv(N, 128));
    k_gemm<2, false, NP1><<<g, TB, 0, stream>>>(bufH, bufT, w0p2, w1p2, b2p, nullptr,
                                                x1, N, NP2);
  }

  // 4) xs = x1 + relu(x@lin1 + b1); z = x1 + relu(x@lin2 + b2)
  {
    dim3 g(NP2 / 16, cdiv(N, 128));
    k_gemm<1, true, KP0><<<g, TB, 0, stream>>>(xpad, nullptr, l1wp, nullptr, l1bp, x1,
                                               xs, N, NP2);
    k_gemm<1, true, KP0><<<g, TB, 0, stream>>>(xpad, nullptr, l2wp, nullptr, l2bp, x1,
                                               zz, N, NP2);
  }

  // 5) link losses
  k_loss<<<cdiv(E, TB), TB, 0, stream>>>(zz, ei, ei + E, accum, E, H2 / 4, NP2, 0);
  k_loss<<<cdiv(E, TB), TB, 0, stream>>>(zz, nei, nei + E, accum, E, H2 / 4, NP2, 1);

  // 6) conv3: out = xs@w0 + tx1(xs)@w1 + b
  k_zero<<<ZB, TB, 0, stream>>>(bufT, (long)N * NP2);
  k_scatter4<<<4096, TB, 0, stream>>>(xs, src, dst, wE, bufT, (long)E, H2 / 4, NP2, NP2);
  k_conv3<<<cdiv(N, TB), TB, 0, stream>>>(xs, bufT, c3w0, c3w1, c3b, out, N, H2 / 4, NP2);

  // 7) r_loss, c1, c2
  k_final<<<1, 64, 0, stream>>>(accum, c1s, c2s, out, N, 1.0f / (float)E);
}